// EMALayer_82454782148609
// MI455X (gfx1250) — compile-verified
//
#include <hip/hip_runtime.h>
#include <hip/hip_bf16.h>

// Problem constants (match reference)
#define Bb 16
#define Ll 256
#define Cc 128
#define Kk 128
#define Rr 128
#define CK 256          // C + K
#define PADROW 260      // CK + 4 pad -> conflict-free WMMA fragment reads
#define NWG 16          // workgroups in recurrent kernel (one cluster)
#define LN_EPS 1e-5f
#define GTILE 1024      // per-WG gumbel tile row: 8 k * 128 c floats

typedef float v2f __attribute__((ext_vector_type(2)));
typedef float v8f __attribute__((ext_vector_type(8)));
typedef unsigned int u32x4 __attribute__((ext_vector_type(4)));
typedef int i32x4 __attribute__((ext_vector_type(4)));
typedef int i32x8 __attribute__((ext_vector_type(8)));

// ---------- helpers ----------
__device__ inline float half_max(float v) {   // reduce over the 16-lane half
    v = fmaxf(v, __shfl_xor(v, 1, 32));
    v = fmaxf(v, __shfl_xor(v, 2, 32));
    v = fmaxf(v, __shfl_xor(v, 4, 32));
    v = fmaxf(v, __shfl_xor(v, 8, 32));
    return v;
}
__device__ inline float half_sum(float v) {
    v += __shfl_xor(v, 1, 32);
    v += __shfl_xor(v, 2, 32);
    v += __shfl_xor(v, 4, 32);
    v += __shfl_xor(v, 8, 32);
    return v;
}

// Inter-WG barrier: cluster barrier fast-path (NOP when not launched as a
// cluster) + global-atomic sense barrier for the flat-dispatch case.
__device__ inline void grid_barrier16(int* ctr, int tid) {
    __threadfence();
    __syncthreads();
    __builtin_amdgcn_s_cluster_barrier();
    if (tid == 0) {
        volatile int* arrive = ctr;
        volatile int* gen    = ctr + 1;
        int g = *gen;
        if (atomicAdd(ctr, 1) == NWG - 1) {
            *arrive = 0;
            __threadfence();
            atomicAdd(ctr + 1, 1);
        } else {
            while (*gen == g) __builtin_amdgcn_s_sleep(1);
        }
    }
    __syncthreads();
    __threadfence();
}

// ---------- kernel 0: barrier init ----------
__global__ void init_kernel(int* ctr) { ctr[0] = 0; ctr[1] = 0; }

// ---------- kernel 1: input LayerNorm (per (b,l) row over C) ----------
__global__ __launch_bounds__(128) void ln_in_kernel(
    const float* __restrict__ x, const float* __restrict__ g,
    const float* __restrict__ bt, float* __restrict__ nx) {
    __shared__ float buf[8];
    const int row = blockIdx.x;           // b*L + l
    const int c = threadIdx.x;
    float v = x[(size_t)row * Cc + c];
    float s = v, s2 = v * v;
    for (int off = 16; off; off >>= 1) {
        s  += __shfl_down(s,  off, 32);
        s2 += __shfl_down(s2, off, 32);
    }
    if ((c & 31) == 0) { buf[c >> 5] = s; buf[4 + (c >> 5)] = s2; }
    __syncthreads();
    float ts  = buf[0] + buf[1] + buf[2] + buf[3];
    float ts2 = buf[4] + buf[5] + buf[6] + buf[7];
    float mean = ts * (1.0f / Cc);
    float var  = ts2 * (1.0f / Cc) - mean * mean;
    float rstd = rsqrtf(var + LN_EPS);
    nx[(size_t)row * Cc + c] = (v - mean) * rstd * g[c] + bt[c];
}

// ---------- kernel 2: recurrent scan ----------
// Grid: 16 WGs (intended as one 16-WG cluster), 256 threads = 8 wave32.
// WG w owns GEMM columns n in [w*1024, w*1024+1024) == k in [w*8, w*8+8).
// Wave v within the WG owns exactly one k = w*8 + v (its 128 c-columns).
// Gumbel slice for step t is a 2-D tile (16 batch rows, stride L*K*C, each
// row 1024 contiguous floats) DMA'd into LDS by the Tensor Data Mover so the
// HBM stream overlaps the WMMA K-loop.
__global__ __launch_bounds__(256) void ema_scan_kernel(
    const float* __restrict__ nx, const float* __restrict__ gumbel,
    const float* __restrict__ thetas,
    const float* __restrict__ ln_h_g, const float* __restrict__ ln_h_b,
    const float* __restrict__ W, const float* __restrict__ bl,
    float* __restrict__ hs, float* ubuf, int* bar) {
    __shared__ float concat[Bb][PADROW];   // A operand: [batch][C | K]
    __shared__ float glds[Bb * GTILE];     // gumbel tile [b][k_local*128 + c]

    const int tid  = threadIdx.x;
    const int wg   = blockIdx.x;           // 0..15
    const int wave = tid >> 5;             // 0..7
    const int lane = tid & 31;
    const int m    = lane & 15;            // column-within-tile / A row
    const int hh   = lane >> 4;            // lane half
    const int kw   = wg * 8 + wave;        // this wave's k node
    const int n_wave = wg * 1024 + wave * 128;
    const float alpha = 1.0f / (1.0f + __expf(-thetas[kw]));

    // h0 = 0
    for (int i = tid; i < Bb * Kk; i += 256)
        concat[i >> 7][Cc + (i & 127)] = 0.0f;
    __syncthreads();

    for (int t = 0; t < Ll; ++t) {
        // ---- TDM: kick off gumbel tile DMA for this step (wave 0 only) ----
        if (wave == 0) {
            const unsigned long long ga =
                (unsigned long long)(uintptr_t)gumbel +
                4ull * (((unsigned long long)t * Kk + (unsigned long long)wg * 8) * Cc);
            const unsigned int lds_base = (unsigned int)(size_t)&glds[0];
            u32x4 g0;
            g0[0] = 1u;                                        // count=1
            g0[1] = lds_base;                                  // lds_addr
            g0[2] = (unsigned int)(ga & 0xffffffffull);        // global_addr lo
            g0[3] = (unsigned int)((ga >> 32) & 0x1ffffffull)  // global_addr hi
                    | (2u << 30);                              // type=2 (image)
            i32x8 g1;
            g1[0] = 0x00020000;                  // data_size=2 (4 bytes)
            g1[1] = (int)(GTILE << 16);          // tensor_dim0 = 1024 (lo16)
            g1[2] = (int)(Bb << 16);             // tensor_dim1 = 16 (lo16)
            g1[3] = (int)(GTILE << 16);          // tile_dim0 = 1024
            g1[4] = Bb;                          // tile_dim1 = 16, tile_dim2 = 0
            g1[5] = Ll * Kk * Cc;                // tensor_dim0_stride lo32
            g1[6] = 0;
            g1[7] = 0;
            const i32x4 z4 = {0, 0, 0, 0};
            const i32x8 z8 = {0, 0, 0, 0, 0, 0, 0, 0};
            __builtin_amdgcn_tensor_load_to_lds(g0, g1, z4, z4, z8, 0);
        }

        // ---- stage nx_t into concat[:, 0:C] ----
        for (int i = tid; i < Bb * Cc; i += 256) {
            int b = i >> 7, c = i & 127;
            concat[b][c] = nx[((size_t)b * Ll + t) * Cc + c];
        }
        __syncthreads();

        // light prefetch of next step's nx rows
        if (t + 1 < Ll && tid < 128) {
            const float* p = &nx[(((size_t)(tid >> 3)) * Ll + (t + 1)) * Cc
                                 + (size_t)(tid & 7) * 16];
            __builtin_prefetch(p, 0, 0);
        }

        // ---- GEMM: logits[b][c] for this wave's k via v_wmma_f32_16x16x4 ----
        v8f acc[8];
        #pragma unroll
        for (int tl = 0; tl < 8; ++tl) acc[tl] = (v8f){0,0,0,0,0,0,0,0};

        for (int kc = 0; kc < CK / 4; ++kc) {        // 64 K-chunks of 4
            const int j0 = kc * 4;
            v2f a;
            a.x = concat[m][j0 + 2 * hh];
            a.y = concat[m][j0 + 2 * hh + 1];
            #pragma unroll
            for (int tl = 0; tl < 8; ++tl) {         // 8 N-tiles (c-chunks)
                const float* wp = &W[(size_t)(n_wave + tl * 16 + m) * CK + j0 + 2 * hh];
                v2f bf;
                bf.x = wp[0];
                bf.y = wp[1];
                acc[tl] = __builtin_amdgcn_wmma_f32_16x16x4_f32(
                    false, a, false, bf, (short)0, acc[tl], false, false);
            }
        }

        // ---- wait for gumbel DMA, make LDS tile visible to all waves ----
        if (wave == 0) __builtin_amdgcn_s_wait_tensorcnt(0);
        __syncthreads();

        // ---- epilogue: + b_logits + gumbel (from LDS) ----
        #pragma unroll
        for (int tl = 0; tl < 8; ++tl) {
            const int c = tl * 16 + m;
            const float bb = bl[(size_t)kw * Cc + c];
            #pragma unroll
            for (int r = 0; r < 8; ++r) {
                const int b = r + 8 * hh;
                const float gn = glds[b * GTILE + wave * 128 + c];
                acc[tl][r] += bb + gn;
            }
        }

        // ---- softmax over c (per b) + sel = <mask, nx_t> ----
        float rmax[8], rsum[8], dot[8];
        #pragma unroll
        for (int r = 0; r < 8; ++r) rmax[r] = -3.4e38f;
        #pragma unroll
        for (int tl = 0; tl < 8; ++tl)
            #pragma unroll
            for (int r = 0; r < 8; ++r) rmax[r] = fmaxf(rmax[r], acc[tl][r]);
        #pragma unroll
        for (int r = 0; r < 8; ++r) rmax[r] = half_max(rmax[r]);

        #pragma unroll
        for (int r = 0; r < 8; ++r) { rsum[r] = 0.0f; dot[r] = 0.0f; }
        #pragma unroll
        for (int tl = 0; tl < 8; ++tl) {
            const int c = tl * 16 + m;
            #pragma unroll
            for (int r = 0; r < 8; ++r) {
                float e = __expf(acc[tl][r] - rmax[r]);
                rsum[r] += e;
                dot[r]  += e * concat[r + 8 * hh][c];
            }
        }
        #pragma unroll
        for (int r = 0; r < 8; ++r) { rsum[r] = half_sum(rsum[r]); dot[r] = half_sum(dot[r]); }

        // ---- EMA pre-LN value u; publish to global exchange buffer ----
        float* ub = ubuf + (size_t)(t & 1) * Bb * Kk;
        if (m == 0) {
            #pragma unroll
            for (int r = 0; r < 8; ++r) {
                const int b = r + 8 * hh;
                const float sel = dot[r] / rsum[r];
                const float u = alpha * sel + (1.0f - alpha) * concat[b][Cc + kw];
                ub[b * Kk + kw] = u;
            }
        }

        grid_barrier16(bar, tid);

        // ---- LayerNorm over K (redundant per WGP): wave -> 2 batches ----
        {
            const int b = 2 * wave + hh;
            volatile float* ubv = ub;
            float uv[8], s = 0.0f, s2 = 0.0f;
            #pragma unroll
            for (int i = 0; i < 8; ++i) {
                const int k = i * 16 + m;
                float u = ubv[b * Kk + k];
                uv[i] = u; s += u; s2 += u * u;
            }
            s = half_sum(s); s2 = half_sum(s2);
            const float mean = s * (1.0f / Kk);
            const float var  = s2 * (1.0f / Kk) - mean * mean;
            const float rstd = rsqrtf(var + LN_EPS);
            #pragma unroll
            for (int i = 0; i < 8; ++i) {
                const int k = i * 16 + m;
                concat[b][Cc + k] = (uv[i] - mean) * rstd * ln_h_g[k] + ln_h_b[k];
            }
        }
        __syncthreads();

        // ---- emit this WG's k-slice of h to the hs buffer ----
        if (tid < 128) {
            const int b = tid >> 3, k2 = wg * 8 + (tid & 7);
            hs[((size_t)b * Ll + t) * Kk + k2] = concat[b][Cc + k2];
        }
        __syncthreads();
    }
}

// ---------- kernel 3: readout GEMM (B*L x K) @ (K x R) ----------
__global__ __launch_bounds__(256) void readout_kernel(
    const float* __restrict__ hs, const float* __restrict__ Wr,
    const float* __restrict__ br, float* __restrict__ out) {
    __shared__ float at[16][Kk + 4];
    const int tid  = threadIdx.x;
    const int wave = tid >> 5;
    const int lane = tid & 31;
    const int m    = lane & 15;
    const int hh   = lane >> 4;
    const int m0   = blockIdx.x * 16;      // 256 M-tiles over B*L = 4096 rows
    const int n0   = wave * 16;            // 8 N-tiles, one per wave

    for (int i = tid; i < 16 * Kk; i += 256)
        at[i >> 7][i & 127] = hs[(size_t)(m0 + (i >> 7)) * Kk + (i & 127)];
    __syncthreads();

    v8f acc = (v8f){0,0,0,0,0,0,0,0};
    for (int kc = 0; kc < Kk / 4; ++kc) {
        const int k0 = kc * 4;
        v2f a;
        a.x = at[m][k0 + 2 * hh];
        a.y = at[m][k0 + 2 * hh + 1];
        const float* wp = &Wr[(size_t)(n0 + m) * Kk + k0 + 2 * hh];
        v2f bf;
        bf.x = wp[0];
        bf.y = wp[1];
        acc = __builtin_amdgcn_wmma_f32_16x16x4_f32(
            false, a, false, bf, (short)0, acc, false, false);
    }
    const float bb = br[n0 + m];
    #pragma unroll
    for (int r = 0; r < 8; ++r)
        out[(size_t)(m0 + r + 8 * hh) * Rr + n0 + m] = acc[r] + bb;
}

// ---------- launcher ----------
extern "C" void kernel_launch(void* const* d_in, const int* in_sizes, int n_in,
                              void* d_out, int out_size, void* d_ws, size_t ws_size,
                              hipStream_t stream) {
    const float* x        = (const float*)d_in[0];
    const float* gumbel   = (const float*)d_in[1];
    const float* thetas   = (const float*)d_in[2];
    const float* ln_in_g  = (const float*)d_in[3];
    const float* ln_in_b  = (const float*)d_in[4];
    const float* ln_h_g   = (const float*)d_in[5];
    const float* ln_h_b   = (const float*)d_in[6];
    const float* W_logits = (const float*)d_in[7];
    const float* b_logits = (const float*)d_in[8];
    const float* W_read   = (const float*)d_in[9];
    const float* b_read   = (const float*)d_in[10];
    float* out = (float*)d_out;

    float* ws   = (float*)d_ws;
    float* nx   = ws;                          // B*L*C = 524288 floats
    float* hs   = ws + 524288;                 // B*L*K = 524288 floats
    float* ubuf = ws + 1048576;                // 2*B*K = 4096 floats
    int*   bar  = (int*)(ws + 1048576 + 4096); // 2 ints

    init_kernel<<<1, 1, 0, stream>>>(bar);
    ln_in_kernel<<<Bb * Ll, Cc, 0, stream>>>(x, ln_in_g, ln_in_b, nx);
    ema_scan_kernel<<<NWG, 256, 0, stream>>>(nx, gumbel, thetas, ln_h_g, ln_h_b,
                                             W_logits, b_logits, hs, ubuf, bar);
    readout_kernel<<<(Bb * Ll) / 16, 256, 0, stream>>>(hs, W_read, b_read, out);
}